// NTMWriteHead_90400471646263
// MI455X (gfx1250) — compile-verified
//
#include <hip/hip_runtime.h>
#include <math.h>

// NTM write head, MI455X (gfx1250, wave32).
// B=4096, C=1024, N=256, M=64. All fp32.
// HBM-bound (~560MB moved => ~24us @ 23.3TB/s). GEMMs use exact-precision
// V_WMMA_F32_16X16X4_F32. Global->LDS staging uses CDNA5 async loads
// (global_load_async_to_lds_b128, ASYNCcnt) so the 64KB tiles bypass VGPRs.
// new_memory is written with non-temporal stores (streamed once, 268MB).

#define B_ 4096
#define C_ 1024
#define N_ 256
#define M_ 64
#define EPS_ 1e-8f

typedef float v2f __attribute__((ext_vector_type(2)));
typedef float v4f __attribute__((ext_vector_type(4)));
typedef float v8f __attribute__((ext_vector_type(8)));

__device__ __forceinline__ float sigmoidf_(float x) {
    return 1.0f / (1.0f + expf(-x));
}
__device__ __forceinline__ float softplusf_(float x) {
    return fmaxf(x, 0.0f) + log1pf(expf(-fabsf(x)));   // stable softplus
}

// CDNA5 async global->LDS copy, GVS mode: uniform 64-bit SGPR base +
// per-lane 32-bit byte offset; LDS dest address per lane (16B aligned).
__device__ __forceinline__ void async_load_b128(unsigned lds_addr, unsigned goff,
                                                unsigned long long sbase) {
    asm volatile("global_load_async_to_lds_b128 %0, %1, %2"
                 :: "v"(lds_addr), "v"(goff), "s"(sbase)
                 : "memory");
}
__device__ __forceinline__ void wait_async_zero() {
#if __has_builtin(__builtin_amdgcn_s_wait_asynccnt)
    __builtin_amdgcn_s_wait_asynccnt(0);
#else
    asm volatile("s_wait_asynccnt 0x0" ::: "memory");
#endif
}
// generic->LDS offset: ISA 10.2, LDS aperture addr[31:0] is the LDS address
__device__ __forceinline__ unsigned lds_off(const void* p) {
    return (unsigned)(unsigned long long)(uintptr_t)p;
}

// ---------------------------------------------------------------------------
// Kernel 1: k = emb@Wk^T+bk ; e = sigmoid(emb@We^T+be) ; a = emb@Wa^T+ba
// One block = 16 batch rows; emb tile (16x1024) async-staged into LDS with
// +4 row pad (stride 1028 floats = 4112B, 16B aligned; A-frag reads hit all
// 64 banks). 4 waves; wave w owns output columns [16w,16w+16) of all three
// 64-wide weight matrices. K loop: 256 steps x 3 v_wmma_f32_16x16x4_f32.
// ---------------------------------------------------------------------------
__global__ void proj_kernel(const float* __restrict__ emb,
                            const float* __restrict__ Wk, const float* __restrict__ bk,
                            const float* __restrict__ We, const float* __restrict__ be,
                            const float* __restrict__ Wa, const float* __restrict__ ba,
                            float* __restrict__ kbuf, float* __restrict__ ebuf,
                            float* __restrict__ abuf)
{
    extern __shared__ float sA[];          // [16][1028]
    const int LDA = 1028;
    const int tid  = threadIdx.x;          // 0..127
    const int row0 = blockIdx.x * 16;      // batch-row tile

    // async-stage emb tile: 4096 b128 transfers, global -> LDS, no VGPR hop
    unsigned long long ebase = (unsigned long long)(uintptr_t)(emb + (size_t)row0 * C_);
    unsigned ldsA = lds_off(sA);
    #pragma unroll
    for (int i = 0; i < 32; ++i) {
        int q  = tid + i * 128;            // float4 index 0..4095
        int r  = q >> 8;                   // 256 float4 per row
        int c4 = q & 255;
        async_load_b128(ldsA + (unsigned)((r * LDA + c4 * 4) * 4),
                        (unsigned)(q * 16), ebase);
    }
    wait_async_zero();
    __syncthreads();

    const int wave = tid >> 5;
    const int lane = tid & 31;
    const int half = lane >> 4;            // 0: K pair {0,1}, 1: K pair {2,3}
    const int r    = lane & 15;            // A: row within tile; B: output col
    const int n0   = wave * 16;            // output-column tile base (0..48)

    const float* wk_row = Wk + (n0 + r) * C_;
    const float* we_row = We + (n0 + r) * C_;
    const float* wa_row = Wa + (n0 + r) * C_;
    const float* arow   = &sA[r * LDA + 2 * half];

    v8f ak = {}, ae = {}, aa = {};
    #pragma unroll 4
    for (int k = 0; k < C_; k += 4) {
        // A frag (16x4 fp32): lane(r,half) holds A[r][k+2h], A[r][k+2h+1]
        v2f af  = *(const v2f*)(arow + k);
        int kk  = k + 2 * half;
        // B frag (4x16 fp32): B[k'][n] = W[n0+r][k'] -> contiguous float2
        v2f bkf = *(const v2f*)(wk_row + kk);
        v2f bef = *(const v2f*)(we_row + kk);
        v2f baf = *(const v2f*)(wa_row + kk);
        ak = __builtin_amdgcn_wmma_f32_16x16x4_f32(false, af, false, bkf, (short)0, ak, false, false);
        ae = __builtin_amdgcn_wmma_f32_16x16x4_f32(false, af, false, bef, (short)0, ae, false, false);
        aa = __builtin_amdgcn_wmma_f32_16x16x4_f32(false, af, false, baf, (short)0, aa, false, false);
    }

    // D layout: VGPR v, lanes<16 -> row v, lanes>=16 -> row v+8; col = lane&15
    const int col = n0 + r;
    const float bk_v = bk[col], be_v = be[col], ba_v = ba[col];
    #pragma unroll
    for (int v = 0; v < 8; ++v) {
        int row = row0 + v + 8 * half;
        kbuf[row * M_ + col] = ak[v] + bk_v;
        ebuf[row * M_ + col] = sigmoidf_(ae[v] + be_v);
        abuf[row * M_ + col] = aa[v] + ba_v;
    }
}

// ---------------------------------------------------------------------------
// Kernel 2: per-row scalar gates: beta (softplus), g (sigmoid),
// s (softmax over 3), gamma (1+softplus). One block per batch row.
// ---------------------------------------------------------------------------
__global__ void gates_kernel(const float* __restrict__ emb,
                             const float* __restrict__ Wbeta, const float* __restrict__ bbeta,
                             const float* __restrict__ Wg,    const float* __restrict__ bg,
                             const float* __restrict__ Ws,    const float* __restrict__ bs,
                             const float* __restrict__ Wgamma,const float* __restrict__ bgamma,
                             float* __restrict__ gates)
{
    __shared__ float red[6 * 128];
    const int b = blockIdx.x, t = threadIdx.x;   // 128 threads
    const float* e = emb + (size_t)b * C_;
    float p0 = 0.f, p1 = 0.f, p2 = 0.f, p3 = 0.f, p4 = 0.f, p5 = 0.f;
    for (int c = t; c < C_; c += 128) {
        float x = e[c];
        p0 += x * Wbeta[c];
        p1 += x * Wg[c];
        p2 += x * Ws[c];
        p3 += x * Ws[C_ + c];
        p4 += x * Ws[2 * C_ + c];
        p5 += x * Wgamma[c];
    }
    red[t] = p0; red[128 + t] = p1; red[256 + t] = p2;
    red[384 + t] = p3; red[512 + t] = p4; red[640 + t] = p5;
    __syncthreads();
    if (t < 6) {
        float s = 0.f;
        for (int i = 0; i < 128; ++i) s += red[t * 128 + i];
        red[t * 128] = s;
    }
    __syncthreads();
    if (t == 0) {
        float beta = softplusf_(red[0] + bbeta[0]);
        float g    = sigmoidf_(red[128] + bg[0]);
        float s0 = red[256] + bs[0], s1 = red[384] + bs[1], s2 = red[512] + bs[2];
        float mx = fmaxf(s0, fmaxf(s1, s2));
        float e0 = expf(s0 - mx), e1 = expf(s1 - mx), e2 = expf(s2 - mx);
        float inv = 1.0f / (e0 + e1 + e2);
        float gamma = 1.0f + softplusf_(red[640] + bgamma[0]);
        float* gp = gates + b * 8;
        gp[0] = beta; gp[1] = g; gp[2] = e0 * inv; gp[3] = e1 * inv;
        gp[4] = e2 * inv; gp[5] = gamma;
    }
}

// ---------------------------------------------------------------------------
// Kernel 3 (fused): content addressing + gate + shift + sharpen + erase/add.
// One block (256 threads) per batch row; memory[b] (256x64, 64KB) async-
// staged once into LDS (row stride 68 floats = 272B, 16B aligned for b128;
// dot phase worst case 2-way bank conflict, irrelevant at HBM bound) and
// reused for the update, halving memory-tensor HBM reads.
// ---------------------------------------------------------------------------
__global__ void addr_update_kernel(const float* __restrict__ memory,
                                   const float* __restrict__ w_prev,
                                   const float* __restrict__ kbuf,
                                   const float* __restrict__ ebuf,
                                   const float* __restrict__ abuf,
                                   const float* __restrict__ gates,
                                   float* __restrict__ w_out,
                                   float* __restrict__ mem_out)
{
    extern __shared__ float smem[];
    const int LDM = 68;                       // padded row stride (floats)
    float* Mld = smem;                        // 256*68
    float* kv  = Mld + 256 * LDM;             // 64
    float* ev  = kv + 64;                     // 64
    float* av  = ev + 64;                     // 64
    float* red = av + 64;                     // 256
    float* wb  = red + 256;                   // 256

    const int b = blockIdx.x, t = threadIdx.x;   // 256 threads

    // async-stage memory[b]: 4096 b128 transfers, global -> LDS directly
    unsigned long long mbase =
        (unsigned long long)(uintptr_t)(memory + (size_t)b * N_ * M_);
    unsigned ldsM = lds_off(Mld);
    #pragma unroll
    for (int i = 0; i < 16; ++i) {
        int q = t + i * 256;                  // float4 index
        int n = q >> 4;
        int m = (q & 15) * 4;
        async_load_b128(ldsM + (unsigned)((n * LDM + m) * 4),
                        (unsigned)(q * 16), mbase);
    }
    if (t < 64) {                             // small per-row vectors via DS
        kv[t] = kbuf[b * M_ + t];
        ev[t] = ebuf[b * M_ + t];
        av[t] = abuf[b * M_ + t];
    }
    const float* gp = gates + b * 8;
    const float beta = gp[0], g = gp[1], s0 = gp[2], s1 = gp[3], s2 = gp[4],
                gamma = gp[5];
    wait_async_zero();
    __syncthreads();

    // per-slot dot + norms (k-norm recomputed per thread from broadcast LDS)
    float dot = 0.f, nrm = 0.f, kn = 0.f;
    const float* mr = &Mld[t * LDM];
    #pragma unroll 8
    for (int m = 0; m < M_; ++m) {
        float mv = mr[m], km = kv[m];
        dot += mv * km; nrm += mv * mv; kn += km * km;
    }
    float cosv = dot / (fmaxf(sqrtf(nrm), EPS_) * fmaxf(sqrtf(kn), EPS_));
    float x = beta * cosv;

    // block softmax over N=256 (one value per thread)
    red[t] = x; __syncthreads();
    for (int s = 128; s > 0; s >>= 1) {
        if (t < s) red[t] = fmaxf(red[t], red[t + s]);
        __syncthreads();
    }
    float xmax = red[0]; __syncthreads();
    float ex = expf(x - xmax);
    red[t] = ex; __syncthreads();
    for (int s = 128; s > 0; s >>= 1) {
        if (t < s) red[t] += red[t + s];
        __syncthreads();
    }
    float wc = ex / red[0]; __syncthreads();

    // interpolation gate
    float wg = g * wc + (1.0f - g) * w_prev[(size_t)b * N_ + t];
    wb[t] = wg; __syncthreads();
    // circular 3-tap shift
    float wsv = s0 * wb[(t + 255) & 255] + s1 * wg + s2 * wb[(t + 1) & 255];
    // sharpen
    float p = powf(wsv, gamma);
    red[t] = p; __syncthreads();              // barrier also fences wb reads
    for (int s = 128; s > 0; s >>= 1) {
        if (t < s) red[t] += red[t + s];
        __syncthreads();
    }
    float w = p / (red[0] + 1e-16f);
    w_out[(size_t)b * N_ + t] = w;
    __syncthreads();
    wb[t] = w; __syncthreads();

    // erase/add update streamed out with non-temporal float4 stores
    v4f* mdst = (v4f*)(mem_out + (size_t)b * N_ * M_);
    #pragma unroll
    for (int i = 0; i < 16; ++i) {
        int q = t + i * 256;
        int n = q >> 4;
        int m = (q & 15) * 4;
        float wn = wb[n];
        const float* s = &Mld[n * LDM + m];
        v4f o;
        o.x = s[0] * (1.0f - wn * ev[m + 0]) + wn * av[m + 0];
        o.y = s[1] * (1.0f - wn * ev[m + 1]) + wn * av[m + 1];
        o.z = s[2] * (1.0f - wn * ev[m + 2]) + wn * av[m + 2];
        o.w = s[3] * (1.0f - wn * ev[m + 3]) + wn * av[m + 3];
        __builtin_nontemporal_store(o, &mdst[q]);
    }
}

// ---------------------------------------------------------------------------
extern "C" void kernel_launch(void* const* d_in, const int* in_sizes, int n_in,
                              void* d_out, int out_size, void* d_ws, size_t ws_size,
                              hipStream_t stream) {
    const float* emb    = (const float*)d_in[0];
    const float* w_prev = (const float*)d_in[1];
    const float* memory = (const float*)d_in[2];
    const float* Wk     = (const float*)d_in[3];
    const float* bk     = (const float*)d_in[4];
    const float* Wbeta  = (const float*)d_in[5];
    const float* bbeta  = (const float*)d_in[6];
    const float* Wg     = (const float*)d_in[7];
    const float* bg     = (const float*)d_in[8];
    const float* Ws     = (const float*)d_in[9];
    const float* bs     = (const float*)d_in[10];
    const float* Wgamma = (const float*)d_in[11];
    const float* bgamma = (const float*)d_in[12];
    const float* We     = (const float*)d_in[13];
    const float* be     = (const float*)d_in[14];
    const float* Wa     = (const float*)d_in[15];
    const float* ba     = (const float*)d_in[16];

    float* out_w   = (float*)d_out;               // [B, N]
    float* out_mem = out_w + (size_t)B_ * N_;     // [B, N, M]

    float* kbuf  = (float*)d_ws;                  // [B, 64]
    float* ebuf  = kbuf + (size_t)B_ * M_;        // [B, 64]
    float* abuf  = ebuf + (size_t)B_ * M_;        // [B, 64]
    float* gates = abuf + (size_t)B_ * M_;        // [B, 8] (beta,g,s0..2,gamma)

    // 1) WMMA fp32 GEMM for k / e / a (async LDS staging of emb tile)
    size_t lds1 = (size_t)16 * 1028 * sizeof(float);   // 65792 B
    proj_kernel<<<B_ / 16, 128, lds1, stream>>>(emb, Wk, bk, We, be, Wa, ba,
                                                kbuf, ebuf, abuf);
    // 2) scalar gates
    gates_kernel<<<B_, 128, 0, stream>>>(emb, Wbeta, bbeta, Wg, bg, Ws, bs,
                                         Wgamma, bgamma, gates);
    // 3) fused addressing + memory update (async LDS staging of memory tile)
    size_t lds3 = (size_t)(256 * 68 + 64 * 3 + 256 + 256) * sizeof(float); // 72448 B
    addr_update_kernel<<<B_, 256, lds3, stream>>>(memory, w_prev, kbuf, ebuf,
                                                  abuf, gates, out_w, out_mem);
}